// MSTGCN_9311489097891
// MI455X (gfx1250) — compile-verified
//
#include <hip/hip_runtime.h>

typedef __attribute__((ext_vector_type(2))) float v2f;
typedef __attribute__((ext_vector_type(8))) float v8f;

#define B_   16
#define N_   2000
#define C0_  12
#define T_   12
#define F_   64
#define P_   12
#define E_   32000
#define M_   (T_*B_)        // 192
#define BNT_ (B_*N_*T_)     // 384000
#define R_   (M_*N_)        // 384000
#define LN_EPS_ 1e-5f

// ---------------------------------------------------------------- transpose x (B,C0,N,T) -> X0 (B,N,C0,T)
__global__ void k_transpose0(const float* __restrict__ x, float* __restrict__ X0) {
  int i = blockIdx.x * 256 + threadIdx.x;            // enumerates (b,n,c,t), t fastest
  int t = i % T_;
  int c = (i / T_) % C0_;
  int n = (i / (T_ * C0_)) % N_;
  int b = i / (T_ * C0_ * N_);
  X0[i] = x[((b * C0_ + c) * N_ + n) * T_ + t];
}

// ---------------------------------------------------------------- deterministic degree + dst-count
__global__ void k_degcnt(const int* __restrict__ src, const int* __restrict__ dst,
                         const float* __restrict__ w, float* __restrict__ deg,
                         int* __restrict__ cnt) {
  int n = blockIdx.x * 256 + threadIdx.x;
  if (n >= N_) return;
  float d = 0.f; int c = 0;
  for (int e = 0; e < E_; ++e) {
    d += (src[e] == n) ? w[e] : 0.f;
    c += (dst[e] == n) ? 1 : 0;
  }
  deg[n] = d; cnt[n] = c;
}

__global__ void k_scan(const int* __restrict__ cnt, int* __restrict__ rowptr) {
  if (threadIdx.x != 0 || blockIdx.x != 0) return;
  int s = 0; rowptr[0] = 0;
  for (int i = 0; i < N_; ++i) { s += cnt[i]; rowptr[i + 1] = s; }
}

// deterministic CSR fill (per-node scan, stable order)
__global__ void k_csrfill(const int* __restrict__ src, const int* __restrict__ dst,
                          const int* __restrict__ rowptr,
                          int* __restrict__ ccol, int* __restrict__ ceid) {
  int n = blockIdx.x * 256 + threadIdx.x;
  if (n >= N_) return;
  int slot = rowptr[n];
  for (int e = 0; e < E_; ++e)
    if (dst[e] == n) { ccol[slot] = src[e]; ceid[slot] = e; ++slot; }
}

// ---------------------------------------------------------------- lambda_max via power iteration (single WG, LDS)
__global__ void __launch_bounds__(1024) k_lambda(
    const float* __restrict__ ew, const float* __restrict__ deg,
    const int* __restrict__ rowptr, const int* __restrict__ ccol,
    const int* __restrict__ ceid, float* __restrict__ lam) {
  __shared__ float sv[N_];
  __shared__ float sy[N_];
  __shared__ float sr[1024];
  int tid = threadIdx.x;
  for (int i = tid; i < N_; i += 1024) sv[i] = rsqrtf((float)N_);
  __syncthreads();
  float rq = 0.f;
  for (int it = 0; it < 128; ++it) {
    // y = L v  (A symmetric: row gather via CSR-by-dst with original weights)
    for (int i = tid; i < N_; i += 1024) {
      float acc = deg[i] * sv[i];
      int e0 = rowptr[i], e1 = rowptr[i + 1];
      for (int j = e0; j < e1; ++j) acc -= ew[ceid[j]] * sv[ccol[j]];
      sy[i] = acc;
    }
    __syncthreads();
    float dn = 0.f, dd = 0.f;
    for (int i = tid; i < N_; i += 1024) { dn += sy[i] * sy[i]; dd += sv[i] * sy[i]; }
    sr[tid] = dn; __syncthreads();
    for (int s = 512; s; s >>= 1) { if (tid < s) sr[tid] += sr[tid + s]; __syncthreads(); }
    float norm2 = sr[0]; __syncthreads();
    sr[tid] = dd; __syncthreads();
    for (int s = 512; s; s >>= 1) { if (tid < s) sr[tid] += sr[tid + s]; __syncthreads(); }
    rq = sr[0]; __syncthreads();
    float inv = rsqrtf(norm2 + 1e-30f);
    for (int i = tid; i < N_; i += 1024) sv[i] = sy[i] * inv;
    __syncthreads();
  }
  if (tid == 0) lam[0] = rq;   // Rayleigh quotient with ||v||=1
}

__global__ void k_coeffs(const float* __restrict__ w, const float* __restrict__ deg,
                         const float* __restrict__ lam, float* __restrict__ lw,
                         float* __restrict__ diag) {
  int i = blockIdx.x * 256 + threadIdx.x;
  float la = lam[0];
  if (i < E_) lw[i] = -2.0f * w[i] / la;
  if (i < N_) diag[i] = 2.0f * deg[i] / la - 1.0f;
}

__global__ void k_gatherval(const float* __restrict__ lw, const int* __restrict__ ceid,
                            float* __restrict__ cval) {
  int i = blockIdx.x * 256 + threadIdx.x;
  if (i < E_) cval[i] = lw[ceid[i]];
}

// ---------------------------------------------------------------- scrambled reshape: X (B,N,C,T) -> G (M,N,C)
__global__ void k_rearrange(const float* __restrict__ xin, float* __restrict__ g, int C) {
  int q = blockIdx.x * 256 + threadIdx.x;            // (m,n2,c2), c2 fastest
  int c2 = q % C;
  int n2 = (q / C) % N_;
  int m  = q / (C * N_);
  int p = n2 * (C * M_) + c2 * M_ + m;               // flat index of (C,B,N,T) tensor
  int c = p / BNT_;
  int r = p % BNT_;
  int b = r / (N_ * T_);
  int rem = r % (N_ * T_);
  int n = rem / T_;
  int t = rem % T_;
  g[q] = xin[((b * N_ + n) * C + c) * T_ + t];
}

// ---------------------------------------------------------------- L_hat apply: out = scale*(diag*in1 + A_hat in1) - in0
__global__ void k_lhat(float* __restrict__ out, const float* __restrict__ in1,
                       const float* __restrict__ in0,
                       const int* __restrict__ rowptr, const int* __restrict__ ccol,
                       const float* __restrict__ cval, const float* __restrict__ diag,
                       float scale, int C) {
  int idx = blockIdx.x * 256 + threadIdx.x;
  int c = idx % C;
  int n = (idx / C) % N_;
  int m = idx / (C * N_);
  float acc = diag[n] * in1[idx];
  int e0 = rowptr[n], e1 = rowptr[n + 1];
  const float* base = in1 + (size_t)m * N_ * C + c;
  for (int j = e0; j < e1; ++j)
    acc = fmaf(cval[j], base[(size_t)ccol[j] * C], acc);
  float r = scale * acc;
  if (in0) r -= in0[idx];
  out[idx] = r;
}

// ---------------------------------------------------------------- Chebyshev GEMM: y = relu(sum_s Tx_s @ W_s + b)
// fp32 WMMA 16x16x4; 8 waves/block, 16 rows/wave, all 64 output cols.
// LDS weight image is pair-interleaved: swz[(k/2)*128 + 2f + (k&1)] = W[k][f]
// so each B fragment (W[ka][f], W[ka+1][f]) is one contiguous ds_load_b64.
__global__ void __launch_bounds__(256) k_cheb(
    const float* __restrict__ t0m, const float* __restrict__ t1m,
    const float* __restrict__ t2m, const float* __restrict__ W,
    const float* __restrict__ bias, float* __restrict__ y, int C) {
  __shared__ float sw[3 * 64 * 64];                  // 48 KB max
  int tid = threadIdx.x;
  int nW = 3 * C * F_;                               // rows RK = 3*C (even), f fastest
  for (int i = tid; i < nW; i += 256) {
    int r = i >> 6, f = i & 63;
    sw[(r >> 1) * 128 + 2 * f + (r & 1)] = W[i];
  }
  __syncthreads();
  int wave = tid >> 5, lane = tid & 31;
  int hi = lane >> 4, lo = lane & 15;
  int r0 = blockIdx.x * 128 + wave * 16;
  int row = r0 + lo;
  v8f acc0 = {0.f,0.f,0.f,0.f,0.f,0.f,0.f,0.f};
  v8f acc1 = acc0, acc2 = acc0, acc3 = acc0;
  const float* mats[3] = { t0m, t1m, t2m };
  for (int s = 0; s < 3; ++s) {
    const float* A = mats[s] + (size_t)row * C;
    __builtin_prefetch(A, 0, 0);
    for (int c0 = 0; c0 < C; c0 += 4) {
      // A 16x4: lanes 0-15 carry K = c0,c0+1 ; lanes 16-31 carry K = c0+2,c0+3 (same rows)
      v2f a = *(const v2f*)(A + c0 + 2 * hi);
      int krow = s * C + c0 + 2 * hi;                // even
      const float* bb = &sw[(krow >> 1) * 128 + 2 * lo];
      v2f b0 = *(const v2f*)(bb +  0);               // cols lo +  0
      v2f b1 = *(const v2f*)(bb + 32);               // cols lo + 16
      v2f b2 = *(const v2f*)(bb + 64);               // cols lo + 32
      v2f b3 = *(const v2f*)(bb + 96);               // cols lo + 48
      acc0 = __builtin_amdgcn_wmma_f32_16x16x4_f32(false, a, false, b0, (short)0, acc0, false, false);
      acc1 = __builtin_amdgcn_wmma_f32_16x16x4_f32(false, a, false, b1, (short)0, acc1, false, false);
      acc2 = __builtin_amdgcn_wmma_f32_16x16x4_f32(false, a, false, b2, (short)0, acc2, false, false);
      acc3 = __builtin_amdgcn_wmma_f32_16x16x4_f32(false, a, false, b3, (short)0, acc3, false, false);
    }
  }
  v8f accs[4] = {acc0, acc1, acc2, acc3};
  for (int ct = 0; ct < 4; ++ct) {
    int colf = ct * 16 + lo;
    float bz = bias[colf];
    for (int v = 0; v < 8; ++v) {
      int rr = r0 + v + 8 * hi;                      // C/D layout: vgpr v, half-wave hi
      float z = accs[ct][v] + bz;
      y[(size_t)rr * F_ + colf] = fmaxf(z, 0.f);
    }
  }
}

// ---------------------------------------------------------------- fused temporal(1x3) + residual(1x1) conv as one WMMA GEMM
// rows = (b,n,t); K = 3*F (temporal im2col from Y(M,N,F)) + C (residual from X(B,N,C,T))
__device__ __forceinline__ float loadA_tr(const float* const yb[3], const bool vq[3],
                                          const float* xb, int k) {
  if (k < 192) {
    int dt = k >> 6, c = k & 63;
    return vq[dt] ? yb[dt][c] : 0.f;
  }
  return xb[(k - 192) * T_];
}

__global__ void __launch_bounds__(256) k_temporal(
    const float* __restrict__ Y, const float* __restrict__ X,
    const float* __restrict__ tw, const float* __restrict__ tb,
    const float* __restrict__ rw, const float* __restrict__ rb,
    float* __restrict__ V, int C) {
  __shared__ float sw[256 * 64];                     // up to 64 KB, pair-interleaved
  int tid = threadIdx.x;
  int Ktot = 192 + C;                                // 204 or 256 (even)
  for (int i = tid; i < Ktot * F_; i += 256) {
    int k = i >> 6, f = i & 63;
    float vv = (k < 192) ? tw[(f * F_ + (k & 63)) * 3 + (k >> 6)]   // tw (F,F,1,3)
                         : rw[f * C + (k - 192)];                   // rw (F,C,1,1)
    sw[(k >> 1) * 128 + 2 * f + (k & 1)] = vv;
  }
  __syncthreads();
  int wave = tid >> 5, lane = tid & 31;
  int hi = lane >> 4, lo = lane & 15;
  int r0 = blockIdx.x * 128 + wave * 16;
  int row = r0 + lo;
  int b = row / (N_ * T_);
  int rm = row % (N_ * T_);
  int n = rm / T_;
  int t = rm % T_;
  const float* yb[3]; bool vq[3];
  for (int dt = 0; dt < 3; ++dt) {
    int ts = t + dt - 1;
    vq[dt] = ((unsigned)ts < (unsigned)T_);
    yb[dt] = Y + ((size_t)(b * T_ + (vq[dt] ? ts : 0)) * N_ + n) * F_;
  }
  const float* xb = X + ((size_t)(b * N_ + n) * C) * T_ + t;
  v8f acc0 = {0.f,0.f,0.f,0.f,0.f,0.f,0.f,0.f};
  v8f acc1 = acc0, acc2 = acc0, acc3 = acc0;
  for (int k0 = 0; k0 < Ktot; k0 += 4) {
    int ka = k0 + 2 * hi;                            // even; k<192 test uniform per step
    v2f a;
    a.x = loadA_tr(yb, vq, xb, ka);
    a.y = loadA_tr(yb, vq, xb, ka + 1);
    const float* bb = &sw[(ka >> 1) * 128 + 2 * lo];
    v2f b0 = *(const v2f*)(bb +  0);
    v2f b1 = *(const v2f*)(bb + 32);
    v2f b2 = *(const v2f*)(bb + 64);
    v2f b3 = *(const v2f*)(bb + 96);
    acc0 = __builtin_amdgcn_wmma_f32_16x16x4_f32(false, a, false, b0, (short)0, acc0, false, false);
    acc1 = __builtin_amdgcn_wmma_f32_16x16x4_f32(false, a, false, b1, (short)0, acc1, false, false);
    acc2 = __builtin_amdgcn_wmma_f32_16x16x4_f32(false, a, false, b2, (short)0, acc2, false, false);
    acc3 = __builtin_amdgcn_wmma_f32_16x16x4_f32(false, a, false, b3, (short)0, acc3, false, false);
  }
  v8f accs[4] = {acc0, acc1, acc2, acc3};
  for (int ct = 0; ct < 4; ++ct) {
    int colf = ct * 16 + lo;
    float bz = tb[colf] + rb[colf];
    for (int v = 0; v < 8; ++v) {
      int rr = r0 + v + 8 * hi;
      float z = accs[ct][v] + bz;
      V[(size_t)rr * F_ + colf] = fmaxf(z, 0.f);     // relu(Yt + Yr)
    }
  }
}

// ---------------------------------------------------------------- LayerNorm over F, emit next-block layout (B,N,F,T)
__global__ void k_ln(const float* __restrict__ V, const float* __restrict__ g,
                     const float* __restrict__ be, float* __restrict__ xout) {
  int row = blockIdx.x * 8 + (threadIdx.x >> 5);     // row = (b*N+n)*T + t
  int lane = threadIdx.x & 31;
  float v0 = V[(size_t)row * F_ + lane];
  float v1 = V[(size_t)row * F_ + 32 + lane];
  float s = v0 + v1, q = v0 * v0 + v1 * v1;
  for (int o = 16; o; o >>= 1) {
    s += __shfl_xor(s, o, 32);
    q += __shfl_xor(q, o, 32);
  }
  float mu = s * (1.f / 64.f);
  float var = q * (1.f / 64.f) - mu * mu;
  float inv = rsqrtf(var + LN_EPS_);
  int t = row % T_;
  int bn = row / T_;
  xout[((size_t)bn * F_ + lane) * T_ + t]        = (v0 - mu) * inv * g[lane]      + be[lane];
  xout[((size_t)bn * F_ + lane + 32) * T_ + t]   = (v1 - mu) * inv * g[lane + 32] + be[lane + 32];
}

// ---------------------------------------------------------------- final projection (B,N,P)
__global__ void k_final(const float* __restrict__ X, const float* __restrict__ fw,
                        const float* __restrict__ fb, float* __restrict__ out) {
  int idx = blockIdx.x * 256 + threadIdx.x;          // (b,n,p), p fastest
  int p = idx % P_;
  int bn = idx / P_;
  const float* xb = X + (size_t)bn * F_ * T_;
  const float* wb = fw + p * T_ * F_;
  float acc = fb[p];
  for (int t = 0; t < T_; ++t)
    for (int f = 0; f < F_; ++f)
      acc = fmaf(xb[f * T_ + t], wb[t * F_ + f], acc);
  out[idx] = acc;
}

// ================================================================ host
extern "C" void kernel_launch(void* const* d_in, const int* in_sizes, int n_in,
                              void* d_out, int out_size, void* d_ws, size_t ws_size,
                              hipStream_t stream) {
  (void)in_sizes; (void)n_in; (void)out_size; (void)ws_size;
  const float* x       = (const float*)d_in[0];
  const int*   ei      = (const int*)  d_in[1];
  const float* ew      = (const float*)d_in[2];
  const float* cheb_w0 = (const float*)d_in[3];
  const float* cheb_b0 = (const float*)d_in[4];
  const float* time_w0 = (const float*)d_in[5];
  const float* time_b0 = (const float*)d_in[6];
  const float* res_w0  = (const float*)d_in[7];
  const float* res_b0  = (const float*)d_in[8];
  const float* ln_g0   = (const float*)d_in[9];
  const float* ln_b0   = (const float*)d_in[10];
  const float* cheb_w  = (const float*)d_in[11];
  const float* cheb_b  = (const float*)d_in[12];
  const float* time_w  = (const float*)d_in[13];
  const float* time_b  = (const float*)d_in[14];
  const float* res_w   = (const float*)d_in[15];
  const float* res_b   = (const float*)d_in[16];
  const float* ln_g    = (const float*)d_in[17];
  const float* ln_b    = (const float*)d_in[18];
  const float* final_w = (const float*)d_in[19];
  const float* final_b = (const float*)d_in[20];
  const int* srcp = ei;
  const int* dstp = ei + E_;

  float* ws = (float*)d_ws;
  const size_t SLAB = (size_t)R_ * F_;               // 24,576,000 floats
  float* s0 = ws;
  float* s1 = s0 + SLAB;
  float* s2 = s1 + SLAB;
  float* s3 = s2 + SLAB;
  float* s4 = s3 + SLAB;
  float* fp = s4 + SLAB;
  float* deg  = fp; fp += N_;
  float* diag = fp; fp += N_;
  float* lw   = fp; fp += E_;
  float* cval = fp; fp += E_;
  float* lam  = fp; fp += 4;
  int* ip     = (int*)fp;
  int* rowptr = ip; ip += (N_ + 1);
  int* cnt    = ip; ip += N_;
  int* ccol   = ip; ip += E_;
  int* ceid   = ip; ip += E_;

  // graph preprocessing (deterministic; reused by all 4 blocks)
  k_transpose0<<<(B_*N_*C0_*T_)/256, 256, 0, stream>>>(x, s0);
  k_degcnt<<<(N_ + 255)/256, 256, 0, stream>>>(srcp, dstp, ew, deg, cnt);
  k_scan<<<1, 1, 0, stream>>>(cnt, rowptr);
  k_csrfill<<<(N_ + 255)/256, 256, 0, stream>>>(srcp, dstp, rowptr, ccol, ceid);
  k_lambda<<<1, 1024, 0, stream>>>(ew, deg, rowptr, ccol, ceid, lam);
  k_coeffs<<<(E_ + 255)/256, 256, 0, stream>>>(ew, deg, lam, lw, diag);
  k_gatherval<<<(E_ + 255)/256, 256, 0, stream>>>(lw, ceid, cval);

  float* xin = s0; float* t0 = s1; float* t1 = s2; float* t2 = s3; float* yb = s4;
  for (int blk = 0; blk < 4; ++blk) {
    int C = (blk == 0) ? C0_ : F_;
    const float* cW = (blk == 0) ? cheb_w0 : cheb_w + (size_t)(blk - 1) * 3 * F_ * F_;
    const float* cB = (blk == 0) ? cheb_b0 : cheb_b + (size_t)(blk - 1) * F_;
    const float* tW = (blk == 0) ? time_w0 : time_w + (size_t)(blk - 1) * F_ * F_ * 3;
    const float* tB = (blk == 0) ? time_b0 : time_b + (size_t)(blk - 1) * F_;
    const float* rW = (blk == 0) ? res_w0  : res_w  + (size_t)(blk - 1) * F_ * F_;
    const float* rB = (blk == 0) ? res_b0  : res_b  + (size_t)(blk - 1) * F_;
    const float* gW = (blk == 0) ? ln_g0   : ln_g   + (size_t)(blk - 1) * F_;
    const float* bW = (blk == 0) ? ln_b0   : ln_b   + (size_t)(blk - 1) * F_;
    int tot = M_ * N_ * C;                           // exact multiple of 256
    k_rearrange<<<tot/256, 256, 0, stream>>>(xin, t0, C);
    k_lhat<<<tot/256, 256, 0, stream>>>(t1, t0, nullptr, rowptr, ccol, cval, diag, 1.0f, C);
    k_lhat<<<tot/256, 256, 0, stream>>>(t2, t1, t0,      rowptr, ccol, cval, diag, 2.0f, C);
    k_cheb<<<R_/128, 256, 0, stream>>>(t0, t1, t2, cW, cB, yb, C);
    k_temporal<<<R_/128, 256, 0, stream>>>(yb, xin, tW, tB, rW, rB, t0, C);   // V into t0 (Tx0 dead)
    k_ln<<<BNT_/8, 256, 0, stream>>>(t0, gW, bW, t1);                         // X' into t1 (Tx1 dead)
    float* tmp = xin; xin = t1; t1 = tmp;
  }
  k_final<<<(B_*N_*P_)/256, 256, 0, stream>>>(xin, final_w, final_b, (float*)d_out);
}